// GCN_More_layer_model_tanh_pyra_35476429865577
// MI455X (gfx1250) — compile-verified
//
#include <hip/hip_runtime.h>

typedef float v2f __attribute__((ext_vector_type(2)));
typedef float v8f __attribute__((ext_vector_type(8)));

#define N_NODES  50000
#define N_EDGES  800000
#define F_IN     128
#define N_GRAPHS 512

// ---------------------------------------------------------------------------
// Degree / pooling-state init (runs every call; ws is poisoned by harness)
// ---------------------------------------------------------------------------
__global__ void init_kernel(float* __restrict__ deg, unsigned* __restrict__ maxk,
                            float* __restrict__ sums, float* __restrict__ counts) {
    int i = blockIdx.x * blockDim.x + threadIdx.x;
    if (i < N_NODES) deg[i] = 1.0f;                 // self-loop: deg = in-degree + 1
    if (i < N_GRAPHS * 32) { maxk[i] = 0u; sums[i] = 0.0f; }
    if (i < N_GRAPHS) counts[i] = 0.0f;
}

__global__ void deg_count_kernel(const int* __restrict__ dst, float* __restrict__ deg) {
    int e = blockIdx.x * blockDim.x + threadIdx.x;
    if (e < N_EDGES) atomicAdd(&deg[dst[e]], 1.0f);
}

__global__ void rsqrt_kernel(float* __restrict__ deg) {
    int i = blockIdx.x * blockDim.x + threadIdx.x;
    if (i < N_NODES) deg[i] = rsqrtf(deg[i]);
}

// ---------------------------------------------------------------------------
// Dense GEMM with fp32 WMMA: G[i][f] = dis[i] * sum_k A[i][k] * W[f][k]
// One wave computes one 16x16 C tile; K-loop in steps of 4.
// A-frag (16x4 f32): lanes 0-15 -> M=lane, K={k,k+1}; lanes 16-31 -> K={k+2,k+3}
// C/D (16x16 f32):   VGPR r -> M = r (lanes 0-15) / r+8 (lanes 16-31), N = lane&15
// ---------------------------------------------------------------------------
__global__ void gemm_wmma_kernel(const float* __restrict__ A,   // [N, K]
                                 const float* __restrict__ W,   // [F, K]
                                 const float* __restrict__ dis, // [N]
                                 float* __restrict__ G,         // [N, F]
                                 int K, int F, int nTiles) {
    int wave = (int)((blockIdx.x * blockDim.x + threadIdx.x) >> 5);
    int lane = threadIdx.x & 31;
    if (wave >= nTiles) return;                     // wave-uniform guard: EXEC stays all-1s
    int nTf = F >> 4;
    int tr  = wave / nTf;
    int tc  = wave - tr * nTf;
    int l15 = lane & 15;
    int hi  = (lane >> 4) & 1;                      // upper half-wave handles K+2,K+3

    const float* arow = A + (size_t)(tr * 16 + l15) * K + 2 * hi;
    const float* brow = W + (size_t)(tc * 16 + l15) * K + 2 * hi;

    v8f acc = {};
    for (int k = 0; k < K; k += 4) {
        v2f a = *(const v2f*)(arow + k);
        v2f b = *(const v2f*)(brow + k);
        acc = __builtin_amdgcn_wmma_f32_16x16x4_f32(
            /*neg_a=*/false, a, /*neg_b=*/false, b,
            /*c_mod=*/(short)0, acc, /*reuse_a=*/false, /*reuse_b=*/false);
    }

    int col = tc * 16 + l15;
    #pragma unroll
    for (int r = 0; r < 8; ++r) {
        int row = tr * 16 + r + 8 * hi;
        G[(size_t)row * F + col] = dis[row] * acc[r];
    }
}

// ---------------------------------------------------------------------------
// Edge scatter: acc[dst] += g[src]   (float4 gather + 4 float atomics)
// Consecutive threads cover consecutive feature chunks of the same edge
// so the gather of g[src] is coalesced within a wave.
// ---------------------------------------------------------------------------
__global__ void edge_scatter_kernel(const int* __restrict__ src, const int* __restrict__ dst,
                                    const float* __restrict__ g, float* __restrict__ acc, int F) {
    long long t = (long long)blockIdx.x * blockDim.x + threadIdx.x;
    int nC = F >> 2;
    long long total = (long long)N_EDGES * nC;
    if (t >= total) return;
    int e = (int)(t / nC);
    int c = (int)(t - (long long)e * nC);
    int s = src[e], d = dst[e];
    float4 v = *(const float4*)(g + (size_t)s * F + 4 * c);
    float* p = acc + (size_t)d * F + 4 * c;
    atomicAdd(p + 0, v.x);
    atomicAdd(p + 1, v.y);
    atomicAdd(p + 2, v.z);
    atomicAdd(p + 3, v.w);
}

// ---------------------------------------------------------------------------
// act[i][f] = tanh(dis[i] * acc[i][f] + b[f])
// ---------------------------------------------------------------------------
__global__ void tanh_kernel(const float* __restrict__ acc, const float* __restrict__ dis,
                            const float* __restrict__ b, float* __restrict__ act, int F) {
    long long i = (long long)blockIdx.x * blockDim.x + threadIdx.x;
    long long total = (long long)N_NODES * F;
    if (i >= total) return;
    int row = (int)(i / F);
    int f   = (int)(i - (long long)row * F);
    act[i] = tanhf(dis[row] * acc[i] + b[f]);
}

// ---------------------------------------------------------------------------
// Pooling: segment max (monotonic uint key) + segment sum + counts
// ---------------------------------------------------------------------------
__device__ __forceinline__ unsigned fkey(float x) {
    unsigned b = __float_as_uint(x);
    return (b & 0x80000000u) ? ~b : (b | 0x80000000u);
}

__global__ void pool_kernel(const float* __restrict__ act, const int* __restrict__ batch,
                            unsigned* __restrict__ maxk, float* __restrict__ sums,
                            float* __restrict__ counts) {
    int t = blockIdx.x * blockDim.x + threadIdx.x;
    if (t >= N_NODES * 32) return;
    int i = t >> 5, f = t & 31;
    float v = act[(size_t)i * 32 + f];
    int gph = batch[i];
    atomicMax(&maxk[gph * 32 + f], fkey(v));
    atomicAdd(&sums[gph * 32 + f], v);
    if (f == 0) atomicAdd(&counts[gph], 1.0f);
}

// ---------------------------------------------------------------------------
// out[g] = Wout[0:32].maxp + Wout[32:64].meanp + bout  (one wave per graph)
// ---------------------------------------------------------------------------
__global__ void out_kernel(const unsigned* __restrict__ maxk, const float* __restrict__ sums,
                           const float* __restrict__ counts, const float* __restrict__ Wout,
                           const float* __restrict__ bout, float* __restrict__ out) {
    int gph  = blockIdx.x;
    int lane = threadIdx.x;
    float cnt = counts[gph];
    unsigned k = maxk[gph * 32 + lane];
    unsigned bb = (k & 0x80000000u) ? (k & 0x7fffffffu) : ~k;
    float maxv  = (cnt > 0.0f) ? __uint_as_float(bb) : 0.0f;
    float meanv = sums[gph * 32 + lane] / fmaxf(cnt, 1.0f);
    float part  = Wout[lane] * maxv + Wout[32 + lane] * meanv;
    #pragma unroll
    for (int o = 16; o > 0; o >>= 1) part += __shfl_down(part, o, 32);
    if (lane == 0) out[gph] = part + bout[0];
}

// ---------------------------------------------------------------------------
extern "C" void kernel_launch(void* const* d_in, const int* in_sizes, int n_in,
                              void* d_out, int out_size, void* d_ws, size_t ws_size,
                              hipStream_t stream) {
    const float* x     = (const float*)d_in[0];
    const int*   ei    = (const int*)d_in[1];        // [2, E] flat, int32
    const int*   src   = ei;
    const int*   dst   = ei + N_EDGES;
    const int*   batch = (const int*)d_in[2];
    const float* W[4]  = {(const float*)d_in[4], (const float*)d_in[6],
                          (const float*)d_in[8], (const float*)d_in[10]};
    const float* B[4]  = {(const float*)d_in[5], (const float*)d_in[7],
                          (const float*)d_in[9], (const float*)d_in[11]};
    const float* Wout  = (const float*)d_in[12];
    const float* bout  = (const float*)d_in[13];
    float* out = (float*)d_out;

    // workspace layout
    float*    R1     = (float*)d_ws;                          // g = dis * (X W^T)
    float*    R2     = R1 + (size_t)N_NODES * 256;            // aggregation accumulator
    float*    R3     = R2 + (size_t)N_NODES * 256;            // activations (layer output)
    float*    dis    = R3 + (size_t)N_NODES * 256;            // deg -> rsqrt(deg)
    unsigned* maxk   = (unsigned*)(dis + N_NODES);            // [G*32]
    float*    sums   = (float*)(maxk + N_GRAPHS * 32);        // [G*32]
    float*    counts = sums + N_GRAPHS * 32;                  // [G]

    init_kernel<<<(N_NODES + 255) / 256, 256, 0, stream>>>(dis, maxk, sums, counts);
    deg_count_kernel<<<(N_EDGES + 255) / 256, 256, 0, stream>>>(dst, dis);
    rsqrt_kernel<<<(N_NODES + 255) / 256, 256, 0, stream>>>(dis);

    const int Ks[4] = {F_IN, 256, 128, 64};
    const int Fs[4] = {256, 128, 64, 32};
    const float* in = x;
    for (int l = 0; l < 4; ++l) {
        int K = Ks[l], F = Fs[l];
        int nTiles = (N_NODES / 16) * (F / 16);
        int blocks = (nTiles + 7) / 8;                         // 8 waves / 256-thread block
        gemm_wmma_kernel<<<blocks, 256, 0, stream>>>(in, W[l], dis, R1, K, F, nTiles);

        hipMemcpyAsync(R2, R1, (size_t)N_NODES * F * sizeof(float),
                       hipMemcpyDeviceToDevice, stream);       // acc = g (self term)

        long long eThreads = (long long)N_EDGES * (F / 4);
        edge_scatter_kernel<<<(unsigned)((eThreads + 255) / 256), 256, 0, stream>>>(
            src, dst, R1, R2, F);

        long long tThreads = (long long)N_NODES * F;
        tanh_kernel<<<(unsigned)((tThreads + 255) / 256), 256, 0, stream>>>(
            R2, dis, B[l], R3, F);
        in = R3;
    }

    pool_kernel<<<(N_NODES * 32 + 255) / 256, 256, 0, stream>>>(R3, batch, maxk, sums, counts);
    out_kernel<<<N_GRAPHS, 32, 0, stream>>>(maxk, sums, counts, Wout, bout, out);
}